// Aggregation_28802050687003
// MI455X (gfx1250) — compile-verified
//
#include <hip/hip_runtime.h>
#include <hip/hip_bf16.h>

typedef __attribute__((ext_vector_type(2))) float v2f;
typedef __attribute__((ext_vector_type(8))) float v8f;

// Column dimension is fixed by the problem (D=128).
#define DIMK 128

__device__ __forceinline__ int lower_bound_i64(const long long* __restrict__ idx,
                                               int n, long long val) {
  int lo = 0, hi = n;
  while (lo < hi) {
    int mid = (lo + hi) >> 1;
    if (idx[mid] < val) lo = mid + 1; else hi = mid;
  }
  return lo;
}

// Materialize starts[n] = lower_bound(index, n) for n in [0, N].
// Thread e writes starts[n] for idx[e-1] < n <= idx[e]; each entry written
// exactly once -> deterministic, no atomics. Total writes = N+1.
__global__ __launch_bounds__(256)
void segment_starts_kernel(const long long* __restrict__ index,
                           int* __restrict__ starts, int E, int N) {
  const int e = blockIdx.x * blockDim.x + threadIdx.x;
  if (e > E) return;
  const long long cur  = (e < E) ? index[e] : (long long)N;
  const long long prev = (e == 0) ? (long long)-1 : index[e - 1];
  for (long long n = prev + 1; n <= cur; ++n) starts[n] = e;
}

// One block (4 wave32) per 16-segment tile. Wave w handles columns [32w, 32w+32).
// Segment-sum expressed as one-hot matmul on V_WMMA_F32_16X16X4_F32:
//   Csum[16 seg x 16 col] += A(one-hot 16x4) x B(4 edge-rows x 16 cols)
//   Ccnt += A x ones   -> per-segment edge counts, same pipe.
__global__ __launch_bounds__(128)
void scatter_mean_wmma(const float* __restrict__ x,
                       const long long* __restrict__ index,
                       float* __restrict__ out,
                       const int* __restrict__ starts,   // may be nullptr
                       int E, int N) {
  const int segBase = blockIdx.x * 16;
  const int wave    = threadIdx.x >> 5;   // 0..3 -> column group
  const int lane    = threadIdx.x & 31;
  const int m       = lane & 15;          // A: M-row / B: N-col within tile
  const int khalf   = (lane >> 4) << 1;   // K pair owned by this lane half: {0,1} or {2,3}
  const int colBase = wave * 32;

  // Contiguous edge range for this 16-segment tile (index is sorted).
  int start, end;
  if (starts) {
    int hi = segBase + 16; if (hi > N) hi = N;
    start = starts[segBase];              // uniform -> scalar loads
    end   = starts[hi];
  } else {
    start = lower_bound_i64(index, E, (long long)segBase);
    end   = lower_bound_i64(index, E, (long long)(segBase + 16));
  }

  const long long segTarget = (long long)(segBase + m);

  v8f csum0 = {};   // cols [colBase, colBase+16)
  v8f csum1 = {};   // cols [colBase+16, colBase+32)
  v8f ccnt  = {};   // per-segment counts (replicated across N)
  v2f bones; bones.x = 1.0f; bones.y = 1.0f;

  // Lane-local streaming pointers: all hot-loop loads become base + imm offset.
  const long long* ip = index + (size_t)(start) + khalf;
  const float*     bp = x + (size_t)(start + khalf) * DIMK + colBase + m;

  int e = start;

  // Main loop: full 4-edge groups entirely inside [start, end) -> no clamps,
  // no range tests, no divergence. The one-hot compare is the only guard
  // (index is sorted, so rows outside this tile can never match segTarget).
  // x is streamed exactly once -> non-temporal loads keep L2 for index/starts.
#pragma unroll 2
  for (; e + 4 <= end; e += 4, ip += 4, bp += 4 * DIMK) {
    const long long ia = ip[0];
    const long long ib = ip[1];
    v2f A;
    A.x = (ia == segTarget) ? 1.0f : 0.0f;
    A.y = (ib == segTarget) ? 1.0f : 0.0f;

    // B: 4 edge-rows x 16 cols; lane (N=m) holds K=khalf / khalf+1.
    v2f B0, B1;
    B0.x = __builtin_nontemporal_load(bp);
    B0.y = __builtin_nontemporal_load(bp + DIMK);
    B1.x = __builtin_nontemporal_load(bp + 16);
    B1.y = __builtin_nontemporal_load(bp + DIMK + 16);

    ccnt  = __builtin_amdgcn_wmma_f32_16x16x4_f32(false, A, false, bones, (short)0, ccnt,  false, false);
    csum0 = __builtin_amdgcn_wmma_f32_16x16x4_f32(false, A, false, B0,    (short)0, csum0, false, false);
    csum1 = __builtin_amdgcn_wmma_f32_16x16x4_f32(false, A, false, B1,    (short)0, csum1, false, false);
  }

  // Branchless clamped tail (at most one partial 4-edge group).
  if (e < end) {
    const int ea = e + khalf;
    const int eb = ea + 1;
    const int ra = (ea < E) ? ea : (E - 1);
    const int rb = (eb < E) ? eb : (E - 1);

    const long long ia = index[ra];   // unconditional loads (clamped, safe)
    const long long ib = index[rb];
    v2f A;
    A.x = ((int)(ea < end) & (int)(ia == segTarget)) ? 1.0f : 0.0f;
    A.y = ((int)(eb < end) & (int)(ib == segTarget)) ? 1.0f : 0.0f;

    const float* r0 = x + (size_t)ra * DIMK + colBase + m;
    const float* r1 = x + (size_t)rb * DIMK + colBase + m;
    v2f B0, B1;
    B0.x = __builtin_nontemporal_load(r0);
    B0.y = __builtin_nontemporal_load(r1);
    B1.x = __builtin_nontemporal_load(r0 + 16);
    B1.y = __builtin_nontemporal_load(r1 + 16);

    ccnt  = __builtin_amdgcn_wmma_f32_16x16x4_f32(false, A, false, bones, (short)0, ccnt,  false, false);
    csum0 = __builtin_amdgcn_wmma_f32_16x16x4_f32(false, A, false, B0,    (short)0, csum0, false, false);
    csum1 = __builtin_amdgcn_wmma_f32_16x16x4_f32(false, A, false, B1,    (short)0, csum1, false, false);
  }

  // C/D layout: VGPR r, lanes 0-15 -> segment row r; lanes 16-31 -> row r+8; lane%16 -> col.
  const int rowOff = (lane < 16) ? 0 : 8;
#pragma unroll
  for (int r = 0; r < 8; ++r) {
    const int row = segBase + r + rowOff;
    if (row < N) {
      const float rec = 1.0f / fmaxf(ccnt[r], 1.0f);   // clamp(1) as in scatter_mean
      __builtin_nontemporal_store(csum0[r] * rec, &out[(size_t)row * DIMK + colBase + m]);
      __builtin_nontemporal_store(csum1[r] * rec, &out[(size_t)row * DIMK + colBase + 16 + m]);
    }
  }
}

extern "C" void kernel_launch(void* const* d_in, const int* in_sizes, int n_in,
                              void* d_out, int out_size, void* d_ws, size_t ws_size,
                              hipStream_t stream) {
  const float*     x     = (const float*)d_in[0];
  const long long* index = (const long long*)d_in[1];
  float*           out   = (float*)d_out;

  const int E = in_sizes[1];          // 640000
  const int N = out_size / DIMK;      // 100000

  // Precompute segment start offsets in workspace if it is large enough;
  // otherwise the main kernel falls back to per-wave binary search.
  int* starts = (ws_size >= (size_t)(N + 1) * sizeof(int)) ? (int*)d_ws : nullptr;
  if (starts) {
    const int threads = 256;
    const int blocks  = (E + 1 + threads - 1) / threads;
    segment_starts_kernel<<<dim3(blocks), dim3(threads), 0, stream>>>(index, starts, E, N);
  }

  const int tiles = (N + 15) / 16;    // one block of 4 wave32 per 16-segment tile
  scatter_mean_wmma<<<dim3(tiles), dim3(128), 0, stream>>>(x, index, out, starts, E, N);
}